// AttentionMechanism_25864293056848
// MI455X (gfx1250) — compile-verified
//
#include <hip/hip_runtime.h>
#include <math.h>

// MI455X / gfx1250, wave32. Fused online-softmax attention pooling.
// out[b,e] = sum_n softmax_n(emb[n,b,:].We) * emb[n,b,e]
// (state_score[b] is constant over n -> cancels in softmax over axis 0)
//
// Streaming: GLOBAL_LOAD_ASYNC_TO_LDS_B128 (ASYNCcnt) double-buffered tiles,
// math: V_WMMA_F32_16X16X4_F32 for both the logit GEMV block and the rank-16
// weighted accumulation.

typedef __attribute__((ext_vector_type(2))) float v2f;
typedef __attribute__((ext_vector_type(8))) float v8f;

#define LSTR 516   // 512 + 4 pad: row*516 % 64 = 4*row -> conflict-free A reads; 2064B rows keep 16B align

__device__ __forceinline__ v8f wmma_f32_4(v2f a, v2f b, v8f c) {
  // D = A(16x4 f32) * B(4x16 f32) + C(16x16 f32)
  return __builtin_amdgcn_wmma_f32_16x16x4_f32(false, a, false, b, (short)0, c,
                                               false, false);
}

// One b128 async DMA per lane: copies 16 bytes from global address gbase + goff to LDS offset lds_off.
__device__ __forceinline__ void async_load_b128(unsigned lds_off,
                                                const float* gbase,
                                                unsigned goff) {
  asm volatile("global_load_async_to_lds_b128 %0, %1, %2"
               :: "v"(lds_off), "v"(goff), "s"(gbase)
               : "memory");
}

// Issue one 16-row x 512-col block (32 KB) as 8 async b128 ops per thread-wave set.
// Per thread: col4 = tid&127 fixed, rows (tid>>7) + 2k, k = 0..7.
__device__ __forceinline__ void issue_block_loads(const float* __restrict__ emb,
                                                  int b, int n0,
                                                  unsigned ldsbase, int tid) {
  #pragma unroll
  for (int k = 0; k < 8; ++k) {
    int i = tid + 256 * k, row = i >> 7, col4 = i & 127;
    unsigned goff = (unsigned)((((n0 + row) * 64 + b) * 512 + col4 * 4)) * 4u;
    unsigned loff = ldsbase + (unsigned)((row * LSTR + col4 * 4) * 4);
    async_load_b128(loff, emb, goff);
  }
}

// One workgroup = one (batch b, n-chunk). 256 threads = 8 waves.
__global__ __launch_bounds__(256)
void attn_stream_kernel(const float* __restrict__ emb,   // (2048, 64, 512) f32
                        const float* __restrict__ W,     // (1024, 1) f32, W_e = W+512
                        float* __restrict__ dst,         // ws partials or final out
                        int rowsPerWG, int finalize)
{
  const int b     = blockIdx.y;
  const int chunk = blockIdx.x;
  const int tid   = threadIdx.x;
  const int w     = tid >> 5;       // wave id 0..7 -> owns e-slice [w*64, w*64+64)
  const int lane  = tid & 31;
  const int half  = lane >> 4;      // lane half selects K pair in WMMA A/B slots
  const int j16   = lane & 15;      // M row (A) / N column (B,C)

  __shared__ __align__(16) float tile[2][16 * LSTR]; // double-buffered 33 KB tiles
  __shared__ float Wlds[512];
  __shared__ float pl[8 * 16];      // per-wave logit partials
  __shared__ float lgs[16];         // reduced logits for current 16-row block

  // LDS byte offsets of the two tile buffers (flat LDS aperture: offset = low 32b).
  const unsigned ldsT0 = (unsigned)(uintptr_t)&tile[0][0];
  const unsigned ldsT1 = (unsigned)(uintptr_t)&tile[1][0];

  // Stage W_e into LDS once (visible after first in-loop barrier).
  Wlds[tid]       = W[512 + tid];
  Wlds[tid + 256] = W[768 + tid];

  float m_run = -INFINITY;
  float s_run = 0.f;
  v8f cacc[4];                      // 4 x 16-col C tiles = this wave's 64 e-values
  #pragma unroll
  for (int t4 = 0; t4 < 4; ++t4) cacc[t4] = (v8f){};

  const int nblk    = rowsPerWG >> 4;
  const int rowBase = chunk * rowsPerWG;

  // Pipeline prologue: DMA block 0 into buffer 0.
  issue_block_loads(emb, b, rowBase, ldsT0, tid);

  for (int blk = 0; blk < nblk; ++blk) {
    const float* tcur = &tile[blk & 1][0];

    if (blk + 1 < nblk) {
      // DMA next block into the other buffer, then wait for the *current*
      // buffer's 8 ops: async loads complete in order, so outstanding<=8
      // means the oldest 8 (current buffer) are done.
      issue_block_loads(emb, b, rowBase + (blk + 1) * 16,
                        (blk & 1) ? ldsT0 : ldsT1, tid);
      asm volatile("s_wait_asynccnt 0x8" ::: "memory");
    } else {
      asm volatile("s_wait_asynccnt 0x0" ::: "memory");
    }
    __syncthreads();  // every wave waited for its own portion -> tile complete

    // ---- Phase L: partial logits for this wave's 64-col slice via WMMA ----
    // A[m,k] = tile[m][k0+k] (lane=j16 is row m), B[k,j] = We[k0+k] (bcast cols).
    v8f cl = (v8f){};
    {
      const int kbase = w * 64 + 2 * half;
      #pragma unroll
      for (int t = 0; t < 16; ++t) {
        int k0 = kbase + 4 * t;
        v2f a  = *(const v2f*)&tcur[j16 * LSTR + k0];  // ds_load_b64, conflict-free
        v2f bb = *(const v2f*)&Wlds[k0];               // same-addr broadcast
        cl = wmma_f32_4(a, bb, cl);
      }
    }
    // C VGPR r holds row M=r (lanes 0-15) / M=8+r (lanes 16-31); cols identical.
    if (lane == 0) {
      #pragma unroll
      for (int r = 0; r < 8; ++r) pl[w * 16 + r] = cl[r];
    }
    if (lane == 16) {
      #pragma unroll
      for (int r = 0; r < 8; ++r) pl[w * 16 + 8 + r] = cl[r];
    }
    __syncthreads();
    if (tid < 16) {                                // reduce 8 wave-partials per row
      float s = 0.f;
      #pragma unroll
      for (int c = 0; c < 8; ++c) s += pl[c * 16 + tid];
      lgs[tid] = s;
    }
    __syncthreads();

    // ---- Online softmax update (identical on every thread) ----
    float lg[16];
    #pragma unroll
    for (int r = 0; r < 16; ++r) lg[r] = lgs[r];
    float mblk = lg[0];
    #pragma unroll
    for (int r = 1; r < 16; ++r) mblk = fmaxf(mblk, lg[r]);
    float m_new = fmaxf(m_run, mblk);
    float scale = __expf(m_run - m_new);           // 0 on first block
    float p[16], psum = 0.f;
    #pragma unroll
    for (int r = 0; r < 16; ++r) { p[r] = __expf(lg[r] - m_new); psum += p[r]; }
    s_run = s_run * scale + psum;
    m_run = m_new;

    // ---- Phase W: acc[e] = acc[e]*scale + sum_r p[r]*tile[r][e] via WMMA ----
    // A[m,k] = p[k0+k] (bcast rows), B[k,j] = tile[k0+k][e0+j].
    #pragma unroll
    for (int t4 = 0; t4 < 4; ++t4) {
      cacc[t4] = cacc[t4] * scale;
      const int e0 = w * 64 + t4 * 16 + j16;
      #pragma unroll
      for (int kk = 0; kk < 4; ++kk) {
        const int r0 = 4 * kk + 2 * half;
        v2f a;
        a.x = half ? p[4 * kk + 2] : p[4 * kk + 0];
        a.y = half ? p[4 * kk + 3] : p[4 * kk + 1];
        v2f bb;
        bb.x = tcur[(r0 + 0) * LSTR + e0];
        bb.y = tcur[(r0 + 1) * LSTR + e0];
        cacc[t4] = wmma_f32_4(a, bb, cacc[t4]);
      }
    }
    __syncthreads();  // WAR: all reads of tcur done before it is DMA-refilled
  }

  // ---- Emit: C rows are replicated, C[0,j] lives in VGPR0 of lanes 0-15 ----
  if (finalize) {
    if (half == 0) {
      float inv = 1.f / s_run;
      #pragma unroll
      for (int t4 = 0; t4 < 4; ++t4) {
        int e = w * 64 + t4 * 16 + j16;
        dst[(size_t)b * 512 + e] = cacc[t4][0] * inv;
      }
    }
  } else {
    float* wsb = dst + ((size_t)(b * gridDim.x + chunk)) * 514;
    if (half == 0) {
      #pragma unroll
      for (int t4 = 0; t4 < 4; ++t4) {
        int e = w * 64 + t4 * 16 + j16;
        wsb[2 + e] = cacc[t4][0];
      }
    }
    if (tid == 0) { wsb[0] = m_run; wsb[1] = s_run; }
  }
}

// Merge nchunk partials per batch with max-rescaling. One block per b.
__global__ __launch_bounds__(256)
void attn_combine_kernel(const float* __restrict__ ws, float* __restrict__ out,
                         int nchunk)
{
  const int b = blockIdx.x, tid = threadIdx.x;
  float M = -INFINITY;
  for (int c = 0; c < nchunk; ++c)
    M = fmaxf(M, ws[((size_t)(b * nchunk + c)) * 514 + 0]);
  float S = 0.f;
  for (int c = 0; c < nchunk; ++c) {
    const float* pb = ws + ((size_t)(b * nchunk + c)) * 514;
    S += pb[1] * __expf(pb[0] - M);
  }
  const float invS = 1.f / S;
  for (int e = tid; e < 512; e += 256) {
    float acc = 0.f;
    for (int c = 0; c < nchunk; ++c) {
      const float* pb = ws + ((size_t)(b * nchunk + c)) * 514;
      acc += pb[2 + e] * __expf(pb[0] - M);
    }
    out[(size_t)b * 512 + e] = acc * invS;
  }
}

extern "C" void kernel_launch(void* const* d_in, const int* in_sizes, int n_in,
                              void* d_out, int out_size, void* d_ws, size_t ws_size,
                              hipStream_t stream) {
  // Inputs: [0] state_tm1 (unused: cancels in softmax), [1] embeddings,
  //         [2] W (W_e = W+512), [3] bias (unused).
  const float* emb = (const float*)d_in[1];
  const float* W   = (const float*)d_in[2];
  float* out       = (float*)d_out;
  float* ws        = (float*)d_ws;
  (void)in_sizes; (void)n_in; (void)out_size;

  const int NCHUNK = 8;
  const size_t need = (size_t)64 * NCHUNK * 514 * sizeof(float);
  if (ws_size >= need) {
    attn_stream_kernel<<<dim3(NCHUNK, 64), 256, 0, stream>>>(emb, W, ws,
                                                             2048 / NCHUNK, 0);
    attn_combine_kernel<<<64, 256, 0, stream>>>(ws, out, NCHUNK);
  } else {
    // Deterministic fallback: one chunk per batch, write d_out directly.
    attn_stream_kernel<<<dim3(1, 64), 256, 0, stream>>>(emb, W, out, 2048, 1);
  }
}